// SFC_CAE_Decoder_15530601743079
// MI455X (gfx1250) — compile-verified
//
#include <hip/hip_runtime.h>
#include <cmath>

// ---------------------------------------------------------------------------
// SFC-CAE decoder for MI455X (gfx1250), fp32 via V_WMMA_F32_16X16X4_F32.
// ---------------------------------------------------------------------------

typedef __attribute__((ext_vector_type(2))) float v2f;
typedef __attribute__((ext_vector_type(8))) float v8f;

#define WMMA_F32_16x16x4(a, b, c) \
  __builtin_amdgcn_wmma_f32_16x16x4_f32(false, (a), false, (b), (short)0, (c), false, false)

#define NPTS 102400
#define BATCH 64

// ---------------------------------------------------------------------------
// GEMM: out[m,n] = tanh( sum_k A[m,k]*W[n,k] + bias[n] )
// A: M x K row-major, W: N x K row-major, out: M x N row-major.
// M,N multiples of 16; K multiple of 4. One 16x16 tile per wave (4 waves/block).
//
// ISA 16x4 f32 A layout: lane L holds row m=(L&15), k = 2*(L>>4)+{0,1} in the
// two fragment dwords; B (4x16) is the mirrored layout (lane = column n).
// ---------------------------------------------------------------------------
__global__ __launch_bounds__(128) void fc_tanh_kernel(
    const float* __restrict__ A, const float* __restrict__ W,
    const float* __restrict__ bias, float* __restrict__ out,
    int M, int N, int K, int ntiles) {
  int wave = threadIdx.x >> 5;
  int lane = threadIdx.x & 31;
  int tile = blockIdx.x * 4 + wave;
  if (tile >= ntiles) return;          // wave-uniform exit: EXEC stays all-1s
  int tn = N >> 4;
  int m0 = (tile / tn) << 4;
  int n0 = (tile % tn) << 4;
  int half = lane >> 4;
  int l15  = lane & 15;

  const float* arow = A + (size_t)(m0 + l15) * K + 2 * half;
  const float* wrow = W + (size_t)(n0 + l15) * K + 2 * half;

  v8f acc = {};
  for (int k = 0; k < K; k += 4) {
    v2f af; af.x = arow[k]; af.y = arow[k + 1];
    v2f bf; bf.x = wrow[k]; bf.y = wrow[k + 1];
    acc = WMMA_F32_16x16x4(af, bf, acc);
  }

  float bn = bias[n0 + l15];
#pragma unroll
  for (int v = 0; v < 8; ++v) {
    int m = m0 + v + 8 * half;         // C/D layout: lane=(col), vgpr v -> row
    out[(size_t)m * N + n0 + l15] = tanhf(acc[v] + bn);
  }
}

// ---------------------------------------------------------------------------
// ConvTranspose1d (stride 4, K=32, pad 16) + tanh, as 4-phase implicit GEMM.
//   out[b,co,4q+r] = tanh( bias[co] +
//       sum_{ci,d in 0..7} x[b,ci,q-3+d] * w[ci,co,4*(7-d)+r] )
// Input element (b,ci,l) lives at xbase[b*sb + (ci*Lin + l)*se]
// (se=2 + sfc offset handles the FC2->conv reshape for layer 1).
// Per wave: one 16-row M tile, 16 cols (Cout padded to 16), 4 phase
// accumulators sharing one A fragment per K-step -> 4 WMMAs per A load.
// ---------------------------------------------------------------------------
__global__ __launch_bounds__(128) void convt_tanh_kernel(
    const float* __restrict__ xbase, int sb, int se,
    const float* __restrict__ w,     // torch layout (Cin, Cout, 32)
    const float* __restrict__ bias,  // (Cout)
    float* __restrict__ out,         // (B, Cout, Lout)
    int Cin, int Cout, int Lin, int Lout, int Q, int Mtotal, int ntiles) {
  int wave = threadIdx.x >> 5;
  int lane = threadIdx.x & 31;
  int tile = blockIdx.x * 4 + wave;
  if (tile >= ntiles) return;          // wave-uniform exit
  int m0   = tile << 4;
  int half = lane >> 4;
  int l15  = lane & 15;
  int K    = Cin << 3;                 // Cin * 8 taps

  // A-fragment row for this lane
  int  mA = m0 + l15;
  bool mv = (mA < Mtotal);
  int  bA = mv ? (mA / Q) : 0;
  int  qA = mv ? (mA - bA * Q) : 0;
  const float* xb = xbase + (size_t)bA * sb;

  bool cok = (l15 < Cout);
  const float* wc = w + (size_t)l15 * 32;   // + (ci*Cout)*32 each ci step

  v8f a0 = {}, a1 = {}, a2 = {}, a3 = {};
  for (int k0 = 0; k0 < K; k0 += 4) {
    int k  = k0 + 2 * half;            // even k -> d in {0,2,4,6}
    int ci = k >> 3;
    int d  = k & 7;
    int i0 = qA - 3 + d;
    int i1 = i0 + 1;
    v2f af;
    af.x = (mv && (unsigned)i0 < (unsigned)Lin) ? xb[(size_t)(ci * Lin + i0) * se] : 0.0f;
    af.y = (mv && (unsigned)i1 < (unsigned)Lin) ? xb[(size_t)(ci * Lin + i1) * se] : 0.0f;

    int kk = 4 * (7 - d);              // tap base for d; d+1 uses kk-4
    const float* wp = wc + (size_t)ci * Cout * 32;
    v2f b0 = {}, b1 = {}, b2 = {}, b3 = {};
    if (cok) {
      b0.x = wp[kk + 0]; b0.y = wp[kk - 4 + 0];
      b1.x = wp[kk + 1]; b1.y = wp[kk - 4 + 1];
      b2.x = wp[kk + 2]; b2.y = wp[kk - 4 + 2];
      b3.x = wp[kk + 3]; b3.y = wp[kk - 4 + 3];
    }
    a0 = WMMA_F32_16x16x4(af, b0, a0);
    a1 = WMMA_F32_16x16x4(af, b1, a1);
    a2 = WMMA_F32_16x16x4(af, b2, a2);
    a3 = WMMA_F32_16x16x4(af, b3, a3);
  }

  if (cok) {
    float bn = bias[l15];
#pragma unroll
    for (int v = 0; v < 8; ++v) {
      int m = m0 + v + 8 * half;
      if (m < Mtotal) {
        int bb = m / Q;
        int qq = m - bb * Q;
        float* op = out + ((size_t)bb * Cout + l15) * Lout + 4 * qq;
        float vals[4] = {a0[v], a1[v], a2[v], a3[v]};
#pragma unroll
        for (int r = 0; r < 4; ++r)
          if (4 * qq + r < Lout) op[r] = tanhf(vals[r] + bn);
      }
    }
  }
}

// ---------------------------------------------------------------------------
// NearestNeighbouring gather per SFC, accumulated in zout (= d_out):
//   first: zout = tanh(t);  else: zout += tanh(t);  last: zout = tanh(zout)
// ---------------------------------------------------------------------------
__global__ __launch_bounds__(256) void nn_accum_kernel(
    const float* __restrict__ y,     // (B, NPTS) conv output for this sfc
    const float* __restrict__ w3,    // (NPTS, 3) [minus, self, plus]
    const float* __restrict__ bb,    // (NPTS)
    const int* __restrict__ ordi, const int* __restrict__ mini,
    const int* __restrict__ plui,
    float* __restrict__ zout, int total, int first, int last) {
  int idx = blockIdx.x * 256 + threadIdx.x;
  if (idx >= total) return;
  int b = idx / NPTS;
  int n = idx - b * NPTS;
  const float* yr = y + (size_t)b * NPTS;
  float t = yr[mini[n]] * w3[3 * n + 0]
          + yr[ordi[n]] * w3[3 * n + 1]
          + yr[plui[n]] * w3[3 * n + 2]
          + bb[n];
  float z = tanhf(t);
  if (!first) z += zout[idx];
  zout[idx] = last ? tanhf(z) : z;
}

// ---------------------------------------------------------------------------
extern "C" void kernel_launch(void* const* d_in, const int* in_sizes, int n_in,
                              void* d_out, int out_size, void* d_ws, size_t ws_size,
                              hipStream_t stream) {
  const float* x      = (const float*)d_in[0];   // (64, 128)
  const float* fc_w1  = (const float*)d_in[1];   // (1024, 128)
  const float* fc_b1  = (const float*)d_in[2];   // (1024)
  const float* fc_w2  = (const float*)d_in[3];   // (12832, 1024)
  const float* fc_b2  = (const float*)d_in[4];   // (12832)
  const float* nn_w   = (const float*)d_in[5];   // (2, NPTS, 3)
  const float* nn_b   = (const float*)d_in[6];   // (2, NPTS)
  const int*   ord_i  = (const int*)d_in[7];     // (2, NPTS)
  const int*   min_i  = (const int*)d_in[8];
  const int*   plu_i  = (const int*)d_in[9];
  const float* conv_w[4] = {(const float*)d_in[10], (const float*)d_in[12],
                            (const float*)d_in[14], (const float*)d_in[16]};
  const float* conv_b[4] = {(const float*)d_in[11], (const float*)d_in[13],
                            (const float*)d_in[15], (const float*)d_in[17]};
  float* out = (float*)d_out;
  float* ws  = (float*)d_ws;

  // workspace arena (floats)
  float* h1 = ws;                                   // 64*1024
  float* h2 = h1 + (size_t)BATCH * 1024;            // 64*12832
  float* t1 = h2 + (size_t)BATCH * 12832;           // 64*16*1601
  float* t2 = t1 + (size_t)BATCH * 16 * 1601;       // 64*16*6401
  float* t3 = t2 + (size_t)BATCH * 16 * 6401;       // 64*8*25601
  float* yb = t3 + (size_t)BATCH * 8 * 25601;       // 64*102400

  // FC1: (64,128)x(1024,128)^T
  {
    int M = BATCH, N = 1024, K = 128;
    int ntiles = (M / 16) * (N / 16);
    fc_tanh_kernel<<<(ntiles + 3) / 4, 128, 0, stream>>>(x, fc_w1, fc_b1, h1, M, N, K, ntiles);
  }
  // FC2: (64,1024)x(12832,1024)^T
  {
    int M = BATCH, N = 12832, K = 1024;
    int ntiles = (M / 16) * (N / 16);
    fc_tanh_kernel<<<(ntiles + 3) / 4, 128, 0, stream>>>(h1, fc_w2, fc_b2, h2, M, N, K, ntiles);
  }

  // conv layer geometry: channels [16,16,16,8,1], lengths 401->1601->6401->25601->102400
  const int   Cin [4] = {16, 16, 16, 8};
  const int   Cout[4] = {16, 16, 8, 1};
  const int   Lin [4] = {401, 1601, 6401, 25601};
  const int   Lout[4] = {1601, 6401, 25601, 102400};
  const int   wstr[4] = {16 * 16 * 32, 16 * 16 * 32, 16 * 8 * 32, 8 * 1 * 32};
  const int   bstr[4] = {16, 16, 8, 1};

  for (int s = 0; s < 2; ++s) {
    const float* src[4] = {h2 + s, t1, t2, t3};          // layer-1 src: reshape view of h2
    const int    sb [4] = {12832, 16 * 1601, 16 * 6401, 8 * 25601};
    const int    se [4] = {2, 1, 1, 1};
    float*       dst[4] = {t1, t2, t3, yb};

    for (int j = 0; j < 4; ++j) {
      int Q      = (Lout[j] + 3) / 4;
      int Mtotal = BATCH * Q;
      int ntiles = (Mtotal + 15) / 16;
      convt_tanh_kernel<<<(ntiles + 3) / 4, 128, 0, stream>>>(
          src[j], sb[j], se[j],
          conv_w[j] + (size_t)s * wstr[j], conv_b[j] + (size_t)s * bstr[j],
          dst[j], Cin[j], Cout[j], Lin[j], Lout[j], Q, Mtotal, ntiles);
    }

    int total = BATCH * NPTS;
    nn_accum_kernel<<<(total + 255) / 256, 256, 0, stream>>>(
        yb, nn_w + (size_t)s * NPTS * 3, nn_b + (size_t)s * NPTS,
        ord_i + (size_t)s * NPTS, min_i + (size_t)s * NPTS, plu_i + (size_t)s * NPTS,
        out, total, s == 0, s == 1);
  }
}